// FaFMIMONet_64_128_128_31963146617593
// MI455X (gfx1250) — compile-verified
//
#include <hip/hip_runtime.h>

#define B_ 4
#define A_ 5
#define C_ 64
#define H_ 128
#define W_ 128
#define HW (H_ * W_)
#define NT 256
#define PPT (HW / NT) /* 64 pixels per thread */

typedef unsigned int uint32x4 __attribute__((ext_vector_type(4)));
typedef unsigned int uint32x8 __attribute__((ext_vector_type(8)));

// masked LDS fetch with clamp (matches reference clip+valid-mask semantics)
__device__ __forceinline__ float sample_i(const float* __restrict__ lds, int y, int x) {
    bool v = (y >= 0) & (y <= H_ - 1) & (x >= 0) & (x <= W_ - 1);
    int yc = y < 0 ? 0 : (y > H_ - 1 ? H_ - 1 : y);
    int xc = x < 0 ? 0 : (x > W_ - 1 ? W_ - 1 : x);
    float val = lds[yc * W_ + xc];
    return v ? val : 0.0f;
}

__global__ __launch_bounds__(NT)
void fafmimo_warp_fuse_kernel(const float* __restrict__ feats,
                              const float* __restrict__ trans,
                              const long long* __restrict__ num_agent,
                              float* __restrict__ out) {
    extern __shared__ char smem_raw[];
    float* s_src = (float*)smem_raw;        // 64 KB source plane (TDM destination)
    float* s_w1  = (float*)smem_raw + HW;   // 64 KB intermediate (rot-warped)

    const int tid = threadIdx.x;
    const int blk = blockIdx.x;
    const int c  = blk % C_;
    const int bi = blk / C_;
    const int i  = bi % A_;
    const int b  = bi / A_;

    const int na = (int)num_agent[b * A_ + 0];

    // self term
    const float* self = feats + ((size_t)(b * A_ + i) * C_ + c) * HW;
    float acc[PPT];
#pragma unroll
    for (int k = 0; k < PPT; ++k) acc[k] = self[tid + k * NT];

    if (i < na) {  // uniform per block
        for (int j = 0; j < A_; ++j) {
            if (j == i || j >= na) continue;  // uniform per block

            // ---- stage feats[b,j,c] plane into LDS with the Tensor Data Mover ----
            // One wave issues a single TDM descriptor: 2D tensor, 128x128 x 4B,
            // tile == tensor, contiguous (stride 128 elements).
            if (tid < 32) {
                const void* gsrc = (const void*)(feats + ((size_t)(b * A_ + j) * C_ + c) * HW);
                unsigned long long ga = (unsigned long long)gsrc;
                unsigned ldsa = (unsigned)(size_t)(void*)s_src;

                uint32x4 g0;
                g0[0] = 1u;                                   // count=1 (valid user D#)
                g0[1] = ldsa;                                 // lds_addr
                g0[2] = (unsigned)ga;                         // global_addr[31:0]
                g0[3] = ((unsigned)(ga >> 32) & 0x01FFFFFFu)  // global_addr[56:32]
                        | (2u << 30);                         // type=2 ("image")

                uint32x8 g1;
                g1[0] = 0x00020000u;      // workgroup_mask=0, data_size=2 (4B), no flags
                g1[1] = (unsigned)W_ << 16;  // tensor_dim0 = 128  (bits[63:48])
                g1[2] = (unsigned)H_ << 16;  // tensor_dim1 = 128  (bits[95:80])
                g1[3] = (unsigned)W_ << 16;  // tile_dim0   = 128  (bits[127:112])
                g1[4] = (unsigned)H_;        // tile_dim1   = 128, tile_dim2 = 0
                g1[5] = (unsigned)W_;        // tensor_dim0_stride = 128 (elements)
                g1[6] = 0u;
                g1[7] = 0u;

                asm volatile("tensor_load_to_lds %0, %1" :: "s"(g0), "s"(g1) : "memory");
                __builtin_amdgcn_s_wait_tensorcnt(0);
            }
            __syncthreads();

            // per-pair transform params (L2-resident broadcast reads)
            const float* T = trans + (size_t)((b * A_ + i) * A_ + j) * 16;
            const float t00 = T[0], t01 = T[1], t03 = T[3];
            const float t10 = T[4], t11 = T[5], t13 = T[7];

            // ---- pass 1: rotation-warp into s_w1 ----
#pragma unroll 4
            for (int k = 0; k < PPT; ++k) {
                const int p  = tid + k * NT;
                const int py = p >> 7, px = p & (W_ - 1);
                const float x = fmaf((float)px, 1.0f / 64.0f, 1.0f / 128.0f - 1.0f);
                const float y = fmaf((float)py, 1.0f / 64.0f, 1.0f / 128.0f - 1.0f);
                const float gx = fmaf(t00, x, t01 * y);
                const float gy = fmaf(t10, x, t11 * y);
                const float ix = fmaf(gx, 64.0f, 63.5f);
                const float iy = fmaf(gy, 64.0f, 63.5f);
                const float ix0f = floorf(ix), iy0f = floorf(iy);
                const float wx1 = ix - ix0f, wy1 = iy - iy0f;
                const float wx0 = 1.0f - wx1, wy0 = 1.0f - wy1;
                const int ix0 = (int)ix0f, iy0 = (int)iy0f;
                const float v00 = sample_i(s_src, iy0,     ix0);
                const float v01 = sample_i(s_src, iy0,     ix0 + 1);
                const float v10 = sample_i(s_src, iy0 + 1, ix0);
                const float v11 = sample_i(s_src, iy0 + 1, ix0 + 1);
                s_w1[p] = wy0 * (wx0 * v00 + wx1 * v01) + wy1 * (wx0 * v10 + wx1 * v11);
            }
            __syncthreads();

            // ---- pass 2: pure translation == constant-weight 2x2 integer stencil ----
            const float dx = 2.0f * t03;
            const float dy = -2.0f * t13;
            const float fdx = floorf(dx), fdy = floorf(dy);
            const float wx1 = dx - fdx, wy1 = dy - fdy;
            const float wx0 = 1.0f - wx1, wy0 = 1.0f - wy1;
            const int ox = (int)fdx, oy = (int)fdy;
            const float w00 = wy0 * wx0, w01 = wy0 * wx1;
            const float w10 = wy1 * wx0, w11 = wy1 * wx1;
#pragma unroll
            for (int k = 0; k < PPT; ++k) {
                const int p  = tid + k * NT;
                const int py = p >> 7, px = p & (W_ - 1);
                const int yy = py + oy, xx = px + ox;
                float v = w00 * sample_i(s_w1, yy,     xx)
                        + w01 * sample_i(s_w1, yy,     xx + 1)
                        + w10 * sample_i(s_w1, yy + 1, xx)
                        + w11 * sample_i(s_w1, yy + 1, xx + 1);
                acc[k] += v;
            }
            __syncthreads();  // protect s_w1/s_src before next j overwrites
        }
    }

    // out[(i*B + b), c, :, :]
    float* o = out + ((size_t)(i * B_ + b) * C_ + c) * HW;
#pragma unroll
    for (int k = 0; k < PPT; ++k) o[tid + k * NT] = acc[k];
}

extern "C" void kernel_launch(void* const* d_in, const int* in_sizes, int n_in,
                              void* d_out, int out_size, void* d_ws, size_t ws_size,
                              hipStream_t stream) {
    const float*     feats = (const float*)d_in[0];
    const float*     trans = (const float*)d_in[1];
    const long long* nat   = (const long long*)d_in[2];
    float* out = (float*)d_out;

    const size_t shmem = 2 * (size_t)HW * sizeof(float);  // 128 KB (< 320 KB WGP LDS)
    (void)hipFuncSetAttribute((const void*)fafmimo_warp_fuse_kernel,
                              hipFuncAttributeMaxDynamicSharedMemorySize, (int)shmem);

    dim3 grid(B_ * A_ * C_);
    dim3 block(NT);
    fafmimo_warp_fuse_kernel<<<grid, block, shmem, stream>>>(feats, trans, nat, out);
}